// AttentionDecoder_22711787061866
// MI455X (gfx1250) — compile-verified
//
#include <hip/hip_runtime.h>
#include <math.h>
#include <stdint.h>

// Problem dims (match reference)
#define Bn   64
#define Sn   128
#define En   512
#define Hn   512
#define An   256
#define Vn   8192
#define Tn   32
#define HEn  (Hn + En)     // 1024
#define G4n  (4 * Hn)      // 2048

typedef __attribute__((ext_vector_type(2))) float v2f;
typedef __attribute__((ext_vector_type(8))) float v8f;

__device__ __forceinline__ v2f ld2(const float* p) {
    return *(const v2f*)p;   // callers guarantee 8B alignment
}

// ---------------------------------------------------------------------------
// Generic f32 WMMA GEMM:  C[M,N] = A[M,K] * B  (+bias | +=C)
//   BT==1 : B is W[N,K] row-major (computes A @ W^T), fragments load as b64
//   BT==0 : B is B[K,N] row-major (column fragments)
//   INIT : 0 = overwrite, 1 = overwrite + bias[n], 2 = accumulate into C
// Block = 128 threads (4 waves). Each wave computes a 32(M) x 64(N) strip:
// two A fragments (rows m0, m0+16) shared against four B fragments -> eight
// chained v_wmma_f32_16x16x4_f32 per K-slice, 6 loads per 8 WMMAs.
// All global accesses are (kernarg base + 32-bit index) so loads stay in the
// global address space with cheap 32-bit index increments (no pointer PHIs).
// Grid: (M/32, N/256). All dims used here divide exactly.
// ---------------------------------------------------------------------------
template <int BT, int INIT>
__global__ __launch_bounds__(128)
void gemm_wmma(const float* __restrict__ Am, const float* __restrict__ Bm,
               const float* __restrict__ bias, float* __restrict__ Cm,
               int M, int N, int K, int lda, int ldc, int c_off) {
    const int wave   = threadIdx.x >> 5;
    const int lane   = threadIdx.x & 31;
    const int l      = lane & 15;   // row (A) / col (B,C) within 16
    const int hiHalf = lane >> 4;   // 0: K=0,1   1: K=2,3
    const int kk     = hiHalf * 2;

    const int m0 = blockIdx.x * 32;
    const int n0 = blockIdx.y * 256 + wave * 64;
    if (m0 >= M || n0 >= N) return;   // wave-uniform -> EXEC stays all-ones

    v8f acc0[4] = {};
    v8f acc1[4] = {};

    // 32-bit element indices (all operand buffers are < 2^32 bytes)
    uint32_t aOff0 = (uint32_t)(m0 + l) * (uint32_t)lda + (uint32_t)kk;
    uint32_t aOff1 = aOff0 + 16u * (uint32_t)lda;
    const uint32_t bStep = BT ? 4u : 4u * (uint32_t)N;
    uint32_t bOff[4];
#pragma unroll
    for (int j = 0; j < 4; ++j) {
        const uint32_t n = (uint32_t)(n0 + j * 16 + l);
        bOff[j] = BT ? (n * (uint32_t)K + (uint32_t)kk)     // row of W
                     : ((uint32_t)kk * (uint32_t)N + n);    // column of B
    }

    // ---- prologue: load K-slice 0 ----
    v2f a0 = ld2(Am + aOff0);
    v2f a1 = ld2(Am + aOff1);
    v2f b[4];
#pragma unroll
    for (int j = 0; j < 4; ++j) {
        if (BT) { b[j] = ld2(Bm + bOff[j]); }
        else    { b[j].x = Bm[bOff[j]]; b[j].y = Bm[bOff[j] + (uint32_t)N]; }
    }

    // ---- steady state: prefetch next slice, multiply current ----
    const int nSlices = K >> 2;
#pragma unroll 2
    for (int it = 0; it < nSlices - 1; ++it) {
        aOff0 += 4u;
        aOff1 += 4u;
        v2f a0n = ld2(Am + aOff0);
        v2f a1n = ld2(Am + aOff1);
        v2f bn[4];
#pragma unroll
        for (int j = 0; j < 4; ++j) {
            bOff[j] += bStep;
            if (BT) { bn[j] = ld2(Bm + bOff[j]); }
            else    { bn[j].x = Bm[bOff[j]]; bn[j].y = Bm[bOff[j] + (uint32_t)N]; }
        }
#pragma unroll
        for (int j = 0; j < 4; ++j) {
            acc0[j] = __builtin_amdgcn_wmma_f32_16x16x4_f32(
                false, a0, false, b[j], (short)0, acc0[j], false, false);
            acc1[j] = __builtin_amdgcn_wmma_f32_16x16x4_f32(
                false, a1, false, b[j], (short)0, acc1[j], false, false);
        }
        a0 = a0n;
        a1 = a1n;
#pragma unroll
        for (int j = 0; j < 4; ++j) b[j] = bn[j];
    }

    // ---- epilogue: final slice ----
#pragma unroll
    for (int j = 0; j < 4; ++j) {
        acc0[j] = __builtin_amdgcn_wmma_f32_16x16x4_f32(
            false, a0, false, b[j], (short)0, acc0[j], false, false);
        acc1[j] = __builtin_amdgcn_wmma_f32_16x16x4_f32(
            false, a1, false, b[j], (short)0, acc1[j], false, false);
    }

    // ---- write-out ----
#pragma unroll
    for (int mt = 0; mt < 2; ++mt) {
#pragma unroll
        for (int j = 0; j < 4; ++j) {
#pragma unroll
            for (int i = 0; i < 8; ++i) {
                const int row = m0 + mt * 16 + i + hiHalf * 8;
                const int col = n0 + j * 16 + l;
                const size_t off = (size_t)row * ldc + (size_t)c_off + col;
                float v = (mt == 0) ? acc0[j][i] : acc1[j][i];
                if (INIT == 1)      v += bias[col];
                else if (INIT == 2) v += Cm[off];
                Cm[off] = v;
            }
        }
    }
}

// ---------------------------------------------------------------------------
// Zero h, c state (re-run every call: launch must be deterministic)
// ---------------------------------------------------------------------------
__global__ void zero_state(float* h, float* c) {
    int idx = blockIdx.x * blockDim.x + threadIdx.x;
    if (idx < Bn * Hn) { h[idx] = 0.0f; c[idx] = 0.0f; }
}

// ---------------------------------------------------------------------------
// Attention step for timestep t. One block (256 threads) per batch element.
//   scores[s] = sum_a tanh(enc_proj[b,s,a] + dec_proj[b,a]) * v_att[a]
//   w = softmax(scores);  ctx[e] = sum_s w[s] * enc_feat[b,s,e]
// Also writes x = [emb[tok] , ctx] and the ctx half of y = [h_new , ctx].
// ---------------------------------------------------------------------------
__global__ __launch_bounds__(256)
void attention_step(const float* __restrict__ enc_proj,
                    const float* __restrict__ enc_feat,
                    const float* __restrict__ dec_proj,
                    const float* __restrict__ v_att,
                    const float* __restrict__ emb,
                    const int*   __restrict__ target,
                    int t, float* __restrict__ x, float* __restrict__ y) {
    const int b   = blockIdx.x;
    const int tid = threadIdx.x;

    __shared__ float s_dp[An];
    __shared__ float s_v[An];
    __shared__ float s_sc[Sn];
    __shared__ float s_part[256];
    __shared__ float s_red[Sn];

    s_dp[tid] = dec_proj[b * An + tid];   // An == 256 == blockDim.x
    s_v[tid]  = v_att[tid];
    __syncthreads();

    // scores: pair of threads per s (each does half the A reduction)
    const int s  = tid >> 1;
    const int hf = tid & 1;
    const float* ep = enc_proj + ((size_t)b * Sn + s) * An + hf * 128;
    float part = 0.0f;
    for (int a = 0; a < 128; ++a) {
        float ev = tanhf(ep[a] + s_dp[hf * 128 + a]);
        part += ev * s_v[hf * 128 + a];
    }
    s_part[tid] = part;
    __syncthreads();
    if (hf == 0) s_sc[s] = s_part[tid] + s_part[tid + 1];
    __syncthreads();

    // softmax over 128 scores
    if (tid < Sn) s_red[tid] = s_sc[tid];
    __syncthreads();
    for (int off = Sn / 2; off >= 1; off >>= 1) {
        if (tid < off) s_red[tid] = fmaxf(s_red[tid], s_red[tid + off]);
        __syncthreads();
    }
    const float mx = s_red[0];
    __syncthreads();
    if (tid < Sn) { float e = __expf(s_sc[tid] - mx); s_sc[tid] = e; s_red[tid] = e; }
    __syncthreads();
    for (int off = Sn / 2; off >= 1; off >>= 1) {
        if (tid < off) s_red[tid] += s_red[tid + off];
        __syncthreads();
    }
    const float inv = 1.0f / s_red[0];
    __syncthreads();
    if (tid < Sn) s_sc[tid] *= inv;
    __syncthreads();

    // context: 2 feature columns per thread (E = 512)
    const float* ef = enc_feat + (size_t)b * Sn * En;
    float acc0 = 0.0f, acc1 = 0.0f;
    for (int s2 = 0; s2 < Sn; ++s2) {
        const float w = s_sc[s2];
        acc0 += w * ef[(size_t)s2 * En + tid];
        acc1 += w * ef[(size_t)s2 * En + tid + 256];
    }
    x[(size_t)b * HEn + Hn + tid]       = acc0;
    x[(size_t)b * HEn + Hn + 256 + tid] = acc1;
    y[(size_t)b * HEn + Hn + tid]       = acc0;
    y[(size_t)b * HEn + Hn + 256 + tid] = acc1;

    // teacher-forced embedding into x[:, 0:H]
    const int tok = (t == 0) ? 0 : target[b * Tn + (t - 1)];
    const float* er = emb + (size_t)tok * Hn;
    x[(size_t)b * HEn + tid]       = er[tid];
    x[(size_t)b * HEn + 256 + tid] = er[tid + 256];
}

// ---------------------------------------------------------------------------
// LSTM cell elementwise update; writes h, c state and y[:, 0:H] = h_new
// ---------------------------------------------------------------------------
__device__ __forceinline__ float sigmoidf(float v) {
    return 1.0f / (1.0f + __expf(-v));
}

__global__ void lstm_step(const float* __restrict__ gates,
                          const float* __restrict__ b_ih,
                          const float* __restrict__ b_hh,
                          float* __restrict__ h, float* __restrict__ c,
                          float* __restrict__ y) {
    const int idx = blockIdx.x * blockDim.x + threadIdx.x;
    if (idx >= Bn * Hn) return;
    const int b = idx / Hn;
    const int j = idx % Hn;
    const float* g = gates + (size_t)b * G4n;
    const float gi = g[j]          + b_ih[j]          + b_hh[j];
    const float gf = g[Hn + j]     + b_ih[Hn + j]     + b_hh[Hn + j];
    const float gg = g[2 * Hn + j] + b_ih[2 * Hn + j] + b_hh[2 * Hn + j];
    const float go = g[3 * Hn + j] + b_ih[3 * Hn + j] + b_hh[3 * Hn + j];
    const float cn = sigmoidf(gf) * c[idx] + sigmoidf(gi) * tanhf(gg);
    const float hn = sigmoidf(go) * tanhf(cn);
    c[idx] = cn;
    h[idx] = hn;
    y[(size_t)b * HEn + j] = hn;
}

// ---------------------------------------------------------------------------
// Host-side orchestration
// ---------------------------------------------------------------------------
extern "C" void kernel_launch(void* const* d_in, const int* in_sizes, int n_in,
                              void* d_out, int out_size, void* d_ws, size_t ws_size,
                              hipStream_t stream) {
    (void)in_sizes; (void)n_in; (void)out_size; (void)ws_size;

    const float* enc_feat = (const float*)d_in[0];   // [B,S,E]
    const int*   target   = (const int*)  d_in[1];   // [B,T]
    const float* emb      = (const float*)d_in[2];   // [V,H]
    const float* W_enc    = (const float*)d_in[3];   // [E,A]
    const float* W_dec    = (const float*)d_in[4];   // [H,A]
    const float* b_att    = (const float*)d_in[5];   // [A]
    const float* v_att    = (const float*)d_in[6];   // [A]
    const float* W_ih     = (const float*)d_in[7];   // [4H, H+E]
    const float* W_hh     = (const float*)d_in[8];   // [4H, H]
    const float* b_ih     = (const float*)d_in[9];   // [4H]
    const float* b_hh     = (const float*)d_in[10];  // [4H]
    const float* W_out    = (const float*)d_in[11];  // [V, H+E]
    const float* b_out    = (const float*)d_in[12];  // [V]
    float* out = (float*)d_out;                      // [B,T,V]

    // Workspace carve-out (all chunks are large power-of-two sizes -> aligned)
    char* ws = (char*)d_ws;
    float* enc_proj = (float*)ws; ws += sizeof(float) * (size_t)Bn * Sn * An; // 8 MB
    float* h        = (float*)ws; ws += sizeof(float) * (size_t)Bn * Hn;
    float* c        = (float*)ws; ws += sizeof(float) * (size_t)Bn * Hn;
    float* dec_proj = (float*)ws; ws += sizeof(float) * (size_t)Bn * An;
    float* x        = (float*)ws; ws += sizeof(float) * (size_t)Bn * HEn;
    float* y        = (float*)ws; ws += sizeof(float) * (size_t)Bn * HEn;
    float* gates    = (float*)ws; ws += sizeof(float) * (size_t)Bn * G4n;

    // h = c = 0
    zero_state<<<(Bn * Hn + 255) / 256, 256, 0, stream>>>(h, c);

    // enc_proj[B*S, A] = enc_feat[B*S, E] @ W_enc[E, A]
    gemm_wmma<0, 0><<<dim3((Bn * Sn) / 32, An / 256), 128, 0, stream>>>(
        enc_feat, W_enc, nullptr, enc_proj,
        Bn * Sn, An, En, /*lda=*/En, /*ldc=*/An, /*c_off=*/0);

    for (int t = 0; t < Tn; ++t) {
        // dec_proj[B, A] = h[B, H] @ W_dec[H, A] + b_att
        gemm_wmma<0, 1><<<dim3(Bn / 32, An / 256), 128, 0, stream>>>(
            h, W_dec, b_att, dec_proj,
            Bn, An, Hn, Hn, An, 0);

        // attention + softmax + context + embedding gather
        attention_step<<<Bn, 256, 0, stream>>>(
            enc_proj, enc_feat, dec_proj, v_att, emb, target, t, x, y);

        // gates[B, 4H] = x[B, H+E] @ W_ih^T
        gemm_wmma<1, 0><<<dim3(Bn / 32, G4n / 256), 128, 0, stream>>>(
            x, W_ih, nullptr, gates,
            Bn, G4n, HEn, HEn, G4n, 0);

        // gates += h[B, H] @ W_hh^T
        gemm_wmma<1, 2><<<dim3(Bn / 32, G4n / 256), 128, 0, stream>>>(
            h, W_hh, nullptr, gates,
            Bn, G4n, Hn, Hn, G4n, 0);

        // LSTM cell -> h, c, y[:,0:H]
        lstm_step<<<(Bn * Hn + 255) / 256, 256, 0, stream>>>(
            gates, b_ih, b_hh, h, c, y);

        // out[:, t, :] = y[B, H+E] @ W_out^T + b_out
        gemm_wmma<1, 1><<<dim3(Bn / 32, Vn / 256), 128, 0, stream>>>(
            y, W_out, b_out, out,
            Bn, Vn, HEn, HEn, /*ldc=*/Tn * Vn, /*c_off=*/t * Vn);
    }
}